// DRaGNet3D_59399397704024
// MI455X (gfx1250) — compile-verified
//
#include <hip/hip_runtime.h>
#include <hip/hip_bf16.h>
#include <math.h>

#define BATCH 2
#define DH 64
#define DW 150
#define NNODE (DH*DW)     // 9600
#define KNN 10
#define C1 128
#define C2 256
#define KC 64             // NetVLAD clusters
#define DESC 256          // output dim
#define VSPLIT 8          // K-split factor for the VLAD GEMM

typedef __attribute__((ext_vector_type(2))) float v2f;
typedef __attribute__((ext_vector_type(8))) float v8f;

// ---------------- helpers ----------------
__device__ inline float breduce(float v, float* sbuf) {
  int t = threadIdx.x;
  sbuf[t] = v; __syncthreads();
  for (int s = blockDim.x >> 1; s > 0; s >>= 1) {
    if (t < s) sbuf[t] += sbuf[t + s];
    __syncthreads();
  }
  float r = sbuf[0]; __syncthreads();
  return r;
}

__device__ inline unsigned long long shfl_xor_u64(unsigned long long v, int m) {
  unsigned int lo = (unsigned int)v, hi = (unsigned int)(v >> 32);
  lo = __shfl_xor(lo, m, 32);
  hi = __shfl_xor(hi, m, 32);
  return ((unsigned long long)hi << 32) | lo;
}

// monotone float->uint map (handles tiny negative d2 from cancellation)
__device__ inline unsigned long long pack_di(float d, int j) {
  unsigned int u = __float_as_uint(d);
  u ^= (u >> 31) ? 0xFFFFFFFFu : 0x80000000u;
  return ((unsigned long long)u << 32) | (unsigned int)j;
}

__device__ inline v8f wmma4(v2f a, v2f b, v8f c) {
  return __builtin_amdgcn_wmma_f32_16x16x4_f32(false, a, false, b, (short)0, c,
                                               false, false);
}

// ---------------- 1. spherical -> cartesian ----------------
__global__ void k_build_pts(const float* __restrict__ depth,
                            float* __restrict__ pts, float* __restrict__ sq) {
  int t = blockIdx.x * blockDim.x + threadIdx.x;
  if (t >= BATCH * NNODE) return;
  int n = t % NNODE;
  int h = n / DW, w = n % DW;
  const float PI = 3.14159265358979323846f;
  float theta = -PI + (2.f * PI) * ((float)w / (float)(DW - 1));
  float phi   = -0.5f * PI + PI * ((float)h / (float)(DH - 1));
  float r = depth[t];
  float cph = cosf(phi), sph = sinf(phi), sth = sinf(theta), cth = cosf(theta);
  float x = r * cph * sth, y = r * sph, z = r * cph * cth;
  pts[(size_t)t * 3 + 0] = x;
  pts[(size_t)t * 3 + 1] = y;
  pts[(size_t)t * 3 + 2] = z;
  sq[t] = x * x + y * y + z * z;
}

// ---------------- 2. kNN (one wave32 per row) ----------------
__global__ void k_knn(const float* __restrict__ pts, const float* __restrict__ sq,
                      int* __restrict__ edges) {
  int wid = blockIdx.x * (blockDim.x >> 5) + (threadIdx.x >> 5);
  int lane = threadIdx.x & 31;
  if (wid >= BATCH * NNODE) return;
  int b = wid / NNODE, i = wid % NNODE;
  const float* pb = pts + (size_t)b * NNODE * 3;
  const float* sb = sq + (size_t)b * NNODE;
  float pix = pb[(size_t)i * 3], piy = pb[(size_t)i * 3 + 1], piz = pb[(size_t)i * 3 + 2];
  float sqi = sb[i];
  float bd[KNN]; int bi[KNN];
  #pragma unroll
  for (int q = 0; q < KNN; q++) { bd[q] = 3.4e38f; bi[q] = 0x7fffffff; }
  for (int j = lane; j < NNODE; j += 32) {
    float dot = pix * pb[(size_t)j * 3] + piy * pb[(size_t)j * 3 + 1] + piz * pb[(size_t)j * 3 + 2];
    float d = sqi + sb[j] - 2.f * dot;
    if (d < bd[KNN - 1]) {
      bd[KNN - 1] = d; bi[KNN - 1] = j;
      #pragma unroll
      for (int q = KNN - 2; q >= 0; --q) {
        if (bd[q + 1] < bd[q]) {
          float td = bd[q]; bd[q] = bd[q + 1]; bd[q + 1] = td;
          int ti = bi[q]; bi[q] = bi[q + 1]; bi[q + 1] = ti;
        }
      }
    }
  }
  // merge: 10 rounds of cross-lane u64 min (keys unique: idx in low bits)
  for (int k = 0; k < KNN; k++) {
    unsigned long long head = pack_di(bd[0], bi[0]);
    unsigned long long best = head;
    #pragma unroll
    for (int o = 16; o > 0; o >>= 1) {
      unsigned long long other = shfl_xor_u64(best, o);
      best = (other < best) ? other : best;
    }
    if (lane == 0) edges[(size_t)wid * KNN + k] = (int)(best & 0xffffffffull);
    if (head == best) {
      #pragma unroll
      for (int q = 0; q < KNN - 1; q++) { bd[q] = bd[q + 1]; bi[q] = bi[q + 1]; }
      bd[KNN - 1] = 3.4e38f; bi[KNN - 1] = 0x7fffffff;
    }
  }
}

// ---------------- 3. GAT1 node features (3->128, K too small for WMMA) -------
__global__ void k_gat1_feat(const float* __restrict__ pts, const float* __restrict__ W,
                            const float* __restrict__ Rw, const float* __restrict__ as_,
                            const float* __restrict__ ad_, float* __restrict__ h,
                            float* __restrict__ r, float* __restrict__ es,
                            float* __restrict__ ed) {
  __shared__ float sbuf[C1];
  int node = blockIdx.x;
  int c = threadIdx.x;
  float x0 = pts[(size_t)node * 3], x1 = pts[(size_t)node * 3 + 1], x2 = pts[(size_t)node * 3 + 2];
  float hv = x0 * W[c] + x1 * W[C1 + c] + x2 * W[2 * C1 + c];
  float rv = x0 * Rw[c] + x1 * Rw[C1 + c] + x2 * Rw[2 * C1 + c];
  h[(size_t)node * C1 + c] = hv;
  r[(size_t)node * C1 + c] = rv;
  float s1 = breduce(hv * as_[c], sbuf);
  float s2 = breduce(hv * ad_[c], sbuf);
  if (c == 0) { es[node] = s1; ed[node] = s2; }
}

// ---------------- 4. GAT attention aggregation (shared by both layers) -------
__global__ void k_gat_agg(const float* __restrict__ h, const float* __restrict__ r,
                          const float* __restrict__ es, const float* __restrict__ ed,
                          const int* __restrict__ edges, const float* __restrict__ bias,
                          float* __restrict__ out, int C, int do_relu) {
  __shared__ float s_alpha[KNN];
  __shared__ int s_nbr[KNN];
  int node = blockIdx.x;
  int b = node / NNODE;
  int c = threadIdx.x;
  if (c < KNN) {
    int j = edges[(size_t)node * KNN + c];
    s_nbr[c] = j;
    float e = es[(size_t)b * NNODE + j] + ed[node];
    s_alpha[c] = (e > 0.f) ? e : 0.2f * e;     // leaky_relu(0.2)
  }
  __syncthreads();
  if (c == 0) {
    float m = s_alpha[0];
    for (int k = 1; k < KNN; k++) m = fmaxf(m, s_alpha[k]);
    float s = 0.f;
    for (int k = 0; k < KNN; k++) { float e = expf(s_alpha[k] - m); s_alpha[k] = e; s += e; }
    float inv = 1.f / s;
    for (int k = 0; k < KNN; k++) s_alpha[k] *= inv;
  }
  __syncthreads();
  float acc = bias[c] + r[(size_t)node * C + c];
  #pragma unroll
  for (int k = 0; k < KNN; k++)
    acc += s_alpha[k] * h[((size_t)b * NNODE + s_nbr[k]) * (size_t)C + c];
  if (do_relu) acc = fmaxf(acc, 0.f);
  out[(size_t)node * C + c] = acc;
}

// ---------------- 5. fp32 WMMA GEMM, row-major A, 16x64 block per wave -------
// C[M,N] = A[M,K] (row-major, leading dim lda) * B[K,N] (row-major ldb).
// One wave owns a 16x64 output block: one v2f (b64) A-fragment load feeds 4
// V_WMMA_F32_16X16X4_F32 issues per k-step (4x A reuse, vectorized A load).
__global__ void k_gemm_nb4(const float* __restrict__ A, const float* __restrict__ B,
                           float* __restrict__ C, int M, int N, int K,
                           int lda, int ldb) {
  int wid = blockIdx.x * (blockDim.x >> 5) + (threadIdx.x >> 5);
  int nblk = N >> 6;
  int total = (M >> 4) * nblk;
  if (wid >= total) return;                  // wave-uniform guard
  int tm = wid / nblk, tb = wid % nblk;
  int m0 = tm << 4, n0 = tb << 6;
  int lane = threadIdx.x & 31;
  int half = lane >> 4, lm = lane & 15;
  v8f acc0 = {0.f,0.f,0.f,0.f,0.f,0.f,0.f,0.f};
  v8f acc1 = acc0, acc2 = acc0, acc3 = acc0;
  // A frag (ISA 7.12.2): lanes 0-15 hold K=kk,kk+1 ; lanes 16-31 K=kk+2,kk+3
  const float* Arow = A + (size_t)(m0 + lm) * lda + 2 * half;   // 8B aligned
  const float* B0 = B + n0 + lm;
  for (int kk = 0; kk < K; kk += 4) {
    v2f a = *(const v2f*)(Arow + kk);
    const float* Bk = B0 + (size_t)(kk + 2 * half) * ldb;
    v2f b0, b1, b2, b3;
    b0.x = Bk[0];  b0.y = Bk[ldb];
    b1.x = Bk[16]; b1.y = Bk[ldb + 16];
    b2.x = Bk[32]; b2.y = Bk[ldb + 32];
    b3.x = Bk[48]; b3.y = Bk[ldb + 48];
    acc0 = wmma4(a, b0, acc0);
    acc1 = wmma4(a, b1, acc1);
    acc2 = wmma4(a, b2, acc2);
    acc3 = wmma4(a, b3, acc3);
  }
  // C/D layout: VGPR v -> row m0 + v + 8*half, col n0 + lm (+16/32/48)
  float* Crow = C + (size_t)(m0 + 8 * half) * N + n0 + lm;
  #pragma unroll
  for (int v = 0; v < 8; v++) {
    Crow[(size_t)v * N]      = acc0[v];
    Crow[(size_t)v * N + 16] = acc1[v];
    Crow[(size_t)v * N + 32] = acc2[v];
    Crow[(size_t)v * N + 48] = acc3[v];
  }
}

// ------- 5b. fp32 WMMA GEMM, transposed-A view + K-split (VLAD accumulation) -
// A element (m,k) at A[m*As_m + k*As_k]. Each wave computes one 16x16 tile over
// a K chunk; partials go to Cp[split][M][N] and are summed in k_intra.
__global__ void k_gemm_at_ksplit(const float* __restrict__ A, const float* __restrict__ B,
                                 float* __restrict__ Cp, int M, int N, int K,
                                 int nsplit, int As_m, int As_k, int ldb) {
  int wid = blockIdx.x * (blockDim.x >> 5) + (threadIdx.x >> 5);
  int tiles_n = N >> 4;
  int tiles = (M >> 4) * tiles_n;
  if (wid >= tiles * nsplit) return;
  int split = wid / tiles;
  int t = wid % tiles;
  int tm = t / tiles_n, tn = t % tiles_n;
  int m0 = tm << 4, n0 = tn << 4;
  int kchunk = K / nsplit;                   // multiple of 4
  int k0 = split * kchunk, k1 = k0 + kchunk;
  int lane = threadIdx.x & 31;
  int half = lane >> 4, lm = lane & 15;
  v8f acc = {0.f,0.f,0.f,0.f,0.f,0.f,0.f,0.f};
  const float* Arow = A + (size_t)(m0 + lm) * As_m;
  const float* Bcol = B + n0 + lm;
  for (int kk = k0; kk < k1; kk += 4) {
    int ka = kk + 2 * half;
    v2f a, bfrag;
    a.x = Arow[(size_t)ka * As_k];
    a.y = Arow[(size_t)(ka + 1) * As_k];
    bfrag.x = Bcol[(size_t)ka * ldb];
    bfrag.y = Bcol[(size_t)(ka + 1) * ldb];
    acc = wmma4(a, bfrag, acc);
  }
  float* Crow = Cp + (size_t)split * M * N + (size_t)(m0 + 8 * half) * N + n0 + lm;
  #pragma unroll
  for (int v = 0; v < 8; v++) Crow[(size_t)v * N] = acc[v];
}

// ---------------- 6. e-dot for GAT2 ----------------
__global__ void k_edot(const float* __restrict__ h, const float* __restrict__ as_,
                       const float* __restrict__ ad_, float* __restrict__ es,
                       float* __restrict__ ed) {
  __shared__ float sbuf[C2];
  int node = blockIdx.x, c = threadIdx.x;
  float hv = h[(size_t)node * C2 + c];
  float s1 = breduce(hv * as_[c], sbuf);
  float s2 = breduce(hv * ad_[c], sbuf);
  if (c == 0) { es[node] = s1; ed[node] = s2; }
}

// ---------------- 7. cluster softmax (64) — one wave per node ----------------
__global__ void k_softmax64(float* __restrict__ logits, const float* __restrict__ cb) {
  int wid = blockIdx.x * (blockDim.x >> 5) + (threadIdx.x >> 5);
  int lane = threadIdx.x & 31;
  if (wid >= BATCH * NNODE) return;
  float* row = logits + (size_t)wid * KC;
  float v0 = row[lane] + cb[lane];
  float v1 = row[lane + 32] + cb[lane + 32];
  float m = fmaxf(v0, v1);
  #pragma unroll
  for (int o = 16; o > 0; o >>= 1) m = fmaxf(m, __shfl_xor(m, o, 32));
  float e0 = expf(v0 - m), e1 = expf(v1 - m);
  float s = e0 + e1;
  #pragma unroll
  for (int o = 16; o > 0; o >>= 1) s += __shfl_xor(s, o, 32);
  row[lane] = e0 / s;
  row[lane + 32] = e1 / s;
}

// ---------------- 8. per-cluster activation sum ----------------
__global__ void k_asum(const float* __restrict__ act, float* __restrict__ asum) {
  __shared__ float sbuf[256];
  int b = blockIdx.x >> 6, kc = blockIdx.x & 63;
  const float* ab = act + (size_t)b * NNODE * KC + kc;
  float s = 0.f;
  for (int n = threadIdx.x; n < NNODE; n += blockDim.x) s += ab[(size_t)n * KC];
  s = breduce(s, sbuf);
  if (threadIdx.x == 0) asum[blockIdx.x] = s;
}

// ------ 9. K-split reduce + center subtract + intra L2 norm per (b,cluster) --
__global__ void k_intra(const float* __restrict__ vladP, const float* __restrict__ asum,
                        const float* __restrict__ cw2, float* __restrict__ vladN,
                        float* __restrict__ gsq) {
  __shared__ float sbuf[256];
  int b = blockIdx.x >> 6, kc = blockIdx.x & 63;
  int d = threadIdx.x;
  float val = 0.f;
  #pragma unroll
  for (int s = 0; s < VSPLIT; s++)
    val += vladP[(((size_t)b * VSPLIT + s) * KC + kc) * DESC + d];
  val -= asum[blockIdx.x] * cw2[(size_t)d * KC + kc];
  float ss = breduce(val * val, sbuf);
  float scale = 1.f / fmaxf(sqrtf(ss), 1e-12f);
  vladN[(size_t)b * KC * DESC + (size_t)kc * DESC + d] = val * scale;
  if (d == 0) gsq[blockIdx.x] = ss * scale * scale;
}

// ---------------- 10. global norm + hidden layer matvec ----------------
__global__ void k_h1(const float* __restrict__ vladN, const float* __restrict__ gsq,
                     const float* __restrict__ h1w, const float* __restrict__ h1b,
                     float* __restrict__ y) {
  __shared__ float sbuf[256];
  __shared__ float s_inv;
  int b = blockIdx.x >> 8;
  int o = blockIdx.x & 255;
  float gp = (threadIdx.x < KC) ? gsq[b * KC + threadIdx.x] : 0.f;
  float gs = breduce(gp, sbuf);
  if (threadIdx.x == 0) s_inv = 1.f / fmaxf(sqrtf(gs), 1e-12f);
  __syncthreads();
  float acc = 0.f;
  for (int f = threadIdx.x; f < DESC * KC; f += blockDim.x) {
    int d = f >> 6, kc = f & 63;   // reference flattens d-major: f = d*64 + kc
    acc += vladN[(size_t)b * KC * DESC + (size_t)kc * DESC + d] * h1w[(size_t)f * DESC + o];
  }
  acc = breduce(acc, sbuf);
  if (threadIdx.x == 0) y[b * DESC + o] = s_inv * acc + h1b[o];
}

// ---------------- 11. context gating ----------------
__global__ void k_gate(const float* __restrict__ y, const float* __restrict__ gw,
                       const float* __restrict__ gb, float* __restrict__ out) {
  int b = blockIdx.x, o = threadIdx.x;
  float g = gb[o];
  for (int c2 = 0; c2 < DESC; c2++) g += y[b * DESC + c2] * gw[(size_t)c2 * DESC + o];
  float yo = y[b * DESC + o];
  out[b * DESC + o] = yo * (1.f / (1.f + expf(-g)));
}

// ---------------- launch ----------------
extern "C" void kernel_launch(void* const* d_in, const int* in_sizes, int n_in,
                              void* d_out, int out_size, void* d_ws, size_t ws_size,
                              hipStream_t stream) {
  // inputs (conv path 0,2..7 is dead code in the reference)
  const float* depth   = (const float*)d_in[1];
  const float* gat1_w  = (const float*)d_in[8];
  const float* gat1_as = (const float*)d_in[9];
  const float* gat1_ad = (const float*)d_in[10];
  const float* gat1_b  = (const float*)d_in[11];
  const float* gat1_rs = (const float*)d_in[12];
  const float* gat2_w  = (const float*)d_in[13];
  const float* gat2_as = (const float*)d_in[14];
  const float* gat2_ad = (const float*)d_in[15];
  const float* gat2_b  = (const float*)d_in[16];
  const float* gat2_rs = (const float*)d_in[17];
  const float* vlad_cw = (const float*)d_in[18];
  const float* vlad_cb = (const float*)d_in[19];
  const float* vlad_c2 = (const float*)d_in[20];
  const float* h1w     = (const float*)d_in[21];
  const float* h1b     = (const float*)d_in[22];
  const float* gate_w  = (const float*)d_in[23];
  const float* gate_b  = (const float*)d_in[24];
  float* out = (float*)d_out;

  // workspace arena (~97 MB)
  float* pts   = (float*)d_ws;                       // 2*9600*3
  float* sq    = pts + (size_t)BATCH * NNODE * 3;    // 2*9600
  int*   edges = (int*)(sq + (size_t)BATCH * NNODE); // 2*9600*10
  float* h1    = (float*)(edges + (size_t)BATCH * NNODE * KNN);
  float* r1    = h1 + (size_t)BATCH * NNODE * C1;
  float* e1s   = r1 + (size_t)BATCH * NNODE * C1;
  float* e1d   = e1s + (size_t)BATCH * NNODE;
  float* g1    = e1d + (size_t)BATCH * NNODE;
  float* h2    = g1 + (size_t)BATCH * NNODE * C1;
  float* r2    = h2 + (size_t)BATCH * NNODE * C2;
  float* e2s   = r2 + (size_t)BATCH * NNODE * C2;
  float* e2d   = e2s + (size_t)BATCH * NNODE;
  float* g2    = e2d + (size_t)BATCH * NNODE;
  float* act   = g2 + (size_t)BATCH * NNODE * C2;    // logits -> softmax in place
  float* asum  = act + (size_t)BATCH * NNODE * KC;   // 128
  float* vladP = asum + BATCH * KC;                  // 2*8*64*256 K-split partials
  float* vladN = vladP + (size_t)BATCH * VSPLIT * KC * DESC;  // 2*64*256
  float* gsq   = vladN + (size_t)BATCH * KC * DESC;  // 128
  float* yv    = gsq + BATCH * KC;                   // 512

  const int nodes = BATCH * NNODE;                   // 19200

  // 1. geometry
  k_build_pts<<<(nodes + 255) / 256, 256, 0, stream>>>(depth, pts, sq);
  // 2. kNN: 1 wave per row, 4 waves per block
  k_knn<<<nodes / 4, 128, 0, stream>>>(pts, sq, edges);
  // 3-4. GAT layer 1 (3 -> 128) + relu
  k_gat1_feat<<<nodes, C1, 0, stream>>>(pts, gat1_w, gat1_rs, gat1_as, gat1_ad,
                                        h1, r1, e1s, e1d);
  k_gat_agg<<<nodes, C1, 0, stream>>>(h1, r1, e1s, e1d, edges, gat1_b, g1, C1, 1);
  // 5. GAT layer 2 GEMMs via f32 WMMA (16x64 blocks): [19200,128] x [128,256]
  {
    int waves = (nodes / 16) * (C2 / 64);            // 4800
    k_gemm_nb4<<<(waves + 7) / 8, 256, 0, stream>>>(g1, gat2_w, h2, nodes, C2, C1, C1, C2);
    k_gemm_nb4<<<(waves + 7) / 8, 256, 0, stream>>>(g1, gat2_rs, r2, nodes, C2, C1, C1, C2);
  }
  k_edot<<<nodes, C2, 0, stream>>>(h2, gat2_as, gat2_ad, e2s, e2d);
  k_gat_agg<<<nodes, C2, 0, stream>>>(h2, r2, e2s, e2d, edges, gat2_b, g2, C2, 0);
  // 6. NetVLAD cluster logits: [19200,256] x [256,64]
  {
    int waves = (nodes / 16) * (KC / 64);            // 1200
    k_gemm_nb4<<<(waves + 7) / 8, 256, 0, stream>>>(g2, vlad_cw, act, nodes, KC, C2, C2, KC);
  }
  k_softmax64<<<nodes / 4, 128, 0, stream>>>(act, vlad_cb);
  k_asum<<<BATCH * KC, 256, 0, stream>>>(act, asum);
  // 7. VLAD accumulation per batch, K-split x8 for occupancy:
  //    vladP[s][kc,d] partial of act^T (64x9600) x g2 (9600x256)
  for (int b = 0; b < BATCH; b++) {
    int waves = (KC / 16) * (DESC / 16) * VSPLIT;    // 512
    k_gemm_at_ksplit<<<(waves + 7) / 8, 256, 0, stream>>>(
        act + (size_t)b * NNODE * KC,                // A^T view: As_m=1, As_k=64
        g2 + (size_t)b * NNODE * C2,
        vladP + (size_t)b * VSPLIT * KC * DESC,
        KC, DESC, NNODE, VSPLIT, 1, KC, C2);
  }
  // 8. split-reduce + center subtract + intra-norm, global norm in hidden layer
  k_intra<<<BATCH * KC, DESC, 0, stream>>>(vladP, asum, vlad_c2, vladN, gsq);
  k_h1<<<BATCH * DESC, 256, 0, stream>>>(vladN, gsq, h1w, h1b, yv);
  k_gate<<<BATCH, DESC, 0, stream>>>(yv, gate_w, gate_b, out);
}